// MPNN_41420664603106
// MI455X (gfx1250) — compile-verified
//
#include <hip/hip_runtime.h>
#include <math.h>

#define NN 15000
#define NE 40000
#define NG 750
#define DD 64
#define KP_TOT 65   // 64 h1 features + 1 bias row

typedef float v2f __attribute__((ext_vector_type(2)));
typedef float v8f __attribute__((ext_vector_type(8)));

__device__ __forceinline__ v8f wmma4(v2f a, v2f b, v8f c) {
    // D = A(16x4 f32) * B(4x16 f32) + C(16x16 f32), full-rate fp32 matrix pipe
    return __builtin_amdgcn_wmma_f32_16x16x4_f32(
        false, a, false, b, (short)0, c, false, false);
}

__device__ __forceinline__ float sigm(float v) { return 1.0f / (1.0f + expf(-v)); }

// ---------------------------------------------------------------------------
// Generic fp32 WMMA GEMM: C[M,N] = act(A[M,K] @ B[K,N] + bias[N])
// block = 256 (8 waves = 2 m-tiles x 4 n-tiles), grid = (ceil(M/32), N/64)
// K must be a multiple of 4; N a multiple of 64.
// ---------------------------------------------------------------------------
template <bool RELU>
__global__ void gemm_wmma(const float* __restrict__ A, const float* __restrict__ B,
                          const float* __restrict__ bias, float* __restrict__ C,
                          int M, int K, int N) {
    const int tid = threadIdx.x, w = tid >> 5, lane = tid & 31;
    const int mt = blockIdx.x * 2 + (w >> 2);
    const int nt = blockIdx.y * 4 + (w & 3);
    const int m = lane & 15, khalf = lane >> 4;
    const int n = nt * 16 + m;
    const int row = mt * 16 + m;
    v8f acc = {};
    for (int k0 = 0; k0 < K; k0 += 4) {
        const int kk = k0 + khalf * 2;
        v2f a;
        if (row < M) { a.x = A[row * K + kk]; a.y = A[row * K + kk + 1]; }
        else         { a.x = 0.f; a.y = 0.f; }
        v2f b;
        b.x = B[kk * N + n];
        b.y = B[(kk + 1) * N + n];
        acc = wmma4(a, b, acc);
    }
    const float bv = bias ? bias[n] : 0.0f;
#pragma unroll
    for (int r = 0; r < 8; ++r) {
        const int rr = mt * 16 + r + 8 * khalf;
        if (rr < M) {
            float v = acc[r] + bv;
            if (RELU) v = fmaxf(v, 0.0f);
            C[rr * N + n] = v;
        }
    }
}

// ---------------------------------------------------------------------------
// Fused NNConv message GEMM:  msg[e,f] = sum_{k',d} hbar[e,k']*g[e,d]*Bsw
// A generated on the fly (rank-1), B pre-swizzled per-lane, scatter via atomics.
// block = 256 (8 waves = 2 edge-pairs x 4 n-tiles), 64 edges/block, grid = NE/64
// ---------------------------------------------------------------------------
__global__ void msg_gemm_scatter(const float* __restrict__ node,
                                 const float* __restrict__ h1,
                                 const float* __restrict__ Bsw,
                                 const int* __restrict__ src,
                                 const int* __restrict__ dst,
                                 float* __restrict__ agg) {
    __shared__ float hb[64][KP_TOT];  // stride 65 -> conflict-free broadcast
    const int tid = threadIdx.x;
    const int e0 = blockIdx.x * 64;
    for (int i = tid; i < 64 * 64; i += 256) {
        const int e = i >> 6, k = i & 63;
        hb[e][k] = h1[(e0 + e) * 64 + k];
    }
    if (tid < 64) hb[tid][64] = 1.0f;  // bias column of hbar
    __syncthreads();

    const int w = tid >> 5, lane = tid & 31;
    const int nt = w & 3, ep = w >> 2;      // ep -> edge-tiles ep*2, ep*2+1
    const int m = lane & 15, khalf = lane >> 4;

    // hoist g fragments (repeat every k') into registers
    const int eA = e0 + (ep * 2 + 0) * 16 + m;
    const int eB = e0 + (ep * 2 + 1) * 16 + m;
    const float* rowA = node + (size_t)src[eA] * 64;
    const float* rowB = node + (size_t)src[eB] * 64;
    v2f gA[16], gB[16];
#pragma unroll
    for (int s = 0; s < 16; ++s) {
        const int d = s * 4 + khalf * 2;
        gA[s] = *(const v2f*)(rowA + d);
        gB[s] = *(const v2f*)(rowB + d);
    }

    v8f accA = {}, accB = {};
    const v2f* B2 = (const v2f*)Bsw;
    for (int kp = 0; kp < KP_TOT; ++kp) {
        const float ha = hb[(ep * 2 + 0) * 16 + m][kp];
        const float hc = hb[(ep * 2 + 1) * 16 + m][kp];
        const v2f* bp = B2 + (size_t)(kp * 16) * 128 + nt * 32 + lane;
#pragma unroll
        for (int s = 0; s < 16; ++s) {
            const v2f b = bp[s * 128];
            v2f a;
            a.x = ha * gA[s].x; a.y = ha * gA[s].y;
            accA = wmma4(a, b, accA);
            a.x = hc * gB[s].x; a.y = hc * gB[s].y;
            accB = wmma4(a, b, accB);
        }
    }

    const int col = nt * 16 + m;
#pragma unroll
    for (int r = 0; r < 8; ++r) {
        const int mm = r + 8 * khalf;
        const int ea = e0 + (ep * 2 + 0) * 16 + mm;
        const int eb = e0 + (ep * 2 + 1) * 16 + mm;
        atomicAdd(&agg[(size_t)dst[ea] * 64 + col], accA[r]);
        atomicAdd(&agg[(size_t)dst[eb] * 64 + col], accB[r]);
    }
}

// ---------------------------------------------------------------------------
// Small helper kernels
// ---------------------------------------------------------------------------
__global__ void zero_kernel(float* p, int n) {
    int i = blockIdx.x * 256 + threadIdx.x;
    if (i < n) p[i] = 0.0f;
}

__global__ void deg_kernel(const int* __restrict__ dst, float* __restrict__ deg) {
    int i = blockIdx.x * 256 + threadIdx.x;
    if (i < NE) atomicAdd(&deg[dst[i]], 1.0f);
}

// Build swizzled B: index = ((kp*16+s)*4+nt)*64 + lane*2 + j  (floats)
// value = kp<64 ? e2_W[kp,d*64+f] : e2_b[d*64+f], d = s*4+(lane/16)*2+j, f = nt*16+lane%16
__global__ void prep_bsw(const float* __restrict__ e2W, const float* __restrict__ e2b,
                         float* __restrict__ Bsw) {
    int i = blockIdx.x * 256 + threadIdx.x;
    if (i >= KP_TOT * 4096) return;
    const int j = i & 1;
    const int lane = (i >> 1) & 31;
    const int nt = (i >> 6) & 3;
    const int s = (i >> 8) & 15;
    const int kp = i >> 12;
    const int d = s * 4 + (lane >> 4) * 2 + j;
    const int f = nt * 16 + (lane & 15);
    Bsw[i] = (kp < 64) ? e2W[kp * 4096 + d * 64 + f] : e2b[d * 64 + f];
}

__global__ void transpose_kernel(const float* __restrict__ W, float* __restrict__ WT,
                                 int R, int Ccols) {
    int i = blockIdx.x * 256 + threadIdx.x;
    if (i < R * Ccols) { int r = i / Ccols, c = i % Ccols; WT[c * R + r] = W[i]; }
}

__global__ void mcalc_kernel(const float* __restrict__ agg, const float* __restrict__ deg,
                             const float* __restrict__ conv_b, float* __restrict__ m) {
    int i = blockIdx.x * 256 + threadIdx.x;
    if (i >= NN * 64) return;
    const int n = i >> 6, j = i & 63;
    const float d = fmaxf(deg[n], 1.0f);
    m[i] = fmaxf(agg[i] / d + conv_b[j], 0.0f);
}

__global__ void gru_gate(const float* __restrict__ gi, const float* __restrict__ gh,
                         float* __restrict__ h) {
    int i = blockIdx.x * 256 + threadIdx.x;
    if (i >= NN * 64) return;
    const int n = i >> 6, j = i & 63;
    const float r = sigm(gi[n * 192 + j] + gh[n * 192 + j]);
    const float z = sigm(gi[n * 192 + 64 + j] + gh[n * 192 + 64 + j]);
    const float nn = tanhf(gi[n * 192 + 128 + j] + r * gh[n * 192 + 128 + j]);
    h[i] = (1.0f - z) * nn + z * h[i];
}

__global__ void lstm_kernel(const float* __restrict__ qstar, float* __restrict__ hl,
                            float* __restrict__ cl, const float* __restrict__ Wih,
                            const float* __restrict__ Whh, const float* __restrict__ bih,
                            const float* __restrict__ bhh) {
    const int g = blockIdx.x, j = threadIdx.x;  // 256 threads
    __shared__ float q[128], hh[64], gg[256];
    if (j < 128) q[j] = qstar[g * 128 + j];
    if (j < 64) hh[j] = hl[g * 64 + j];
    __syncthreads();
    float s = bih[j] + bhh[j];
    const float* wi = Wih + j * 128;
    for (int k = 0; k < 128; ++k) s += q[k] * wi[k];
    const float* wh = Whh + j * 64;
    for (int k = 0; k < 64; ++k) s += hh[k] * wh[k];
    gg[j] = s;
    __syncthreads();
    if (j < 64) {
        const float i_ = sigm(gg[j]);
        const float f_ = sigm(gg[64 + j]);
        const float g_ = tanhf(gg[128 + j]);
        const float o_ = sigm(gg[192 + j]);
        const float c = f_ * cl[g * 64 + j] + i_ * g_;
        cl[g * 64 + j] = c;
        hl[g * 64 + j] = o_ * tanhf(c);
    }
}

__global__ void attn_kernel(const float* __restrict__ node, const int* __restrict__ batch,
                            const float* __restrict__ hl, float* __restrict__ qstar,
                            float* __restrict__ ebuf) {
    const int g = blockIdx.x, tid = threadIdx.x;  // 256 threads
    __shared__ float q[64];
    __shared__ float red[256];
    __shared__ int bounds[2];
    if (tid < 64) q[tid] = hl[g * 64 + tid];
    if (tid == 0) {
        int lo = 0, hi = NN;
        while (lo < hi) { int mid = (lo + hi) >> 1; if (batch[mid] < g) lo = mid + 1; else hi = mid; }
        bounds[0] = lo;
        int lo2 = lo, hi2 = NN;
        while (lo2 < hi2) { int mid = (lo2 + hi2) >> 1; if (batch[mid] < g + 1) lo2 = mid + 1; else hi2 = mid; }
        bounds[1] = lo2;
    }
    __syncthreads();
    const int lo = bounds[0], hi = bounds[1];
    float lmax = -INFINITY;
    for (int i = lo + tid; i < hi; i += 256) {
        const float* row = node + (size_t)i * 64;
        float e = 0.0f;
        for (int k = 0; k < 64; ++k) e += row[k] * q[k];
        ebuf[i] = e;
        lmax = fmaxf(lmax, e);
    }
    red[tid] = lmax; __syncthreads();
    for (int s = 128; s > 0; s >>= 1) {
        if (tid < s) red[tid] = fmaxf(red[tid], red[tid + s]);
        __syncthreads();
    }
    const float mx = red[0]; __syncthreads();
    float lsum = 0.0f;
    for (int i = lo + tid; i < hi; i += 256) lsum += expf(ebuf[i] - mx);
    red[tid] = lsum; __syncthreads();
    for (int s = 128; s > 0; s >>= 1) {
        if (tid < s) red[tid] += red[tid + s];
        __syncthreads();
    }
    const float ssum = red[0];
    const float inv = (ssum > 0.0f) ? 1.0f / ssum : 0.0f;
    const int j = tid & 63, grp = tid >> 6;
    float racc = 0.0f;
    for (int i = lo + grp; i < hi; i += 4)
        racc += expf(ebuf[i] - mx) * inv * node[(size_t)i * 64 + j];
    __syncthreads();
    red[tid] = racc; __syncthreads();
    if (tid < 64) {
        qstar[g * 128 + tid] = q[tid];
        qstar[g * 128 + 64 + tid] = red[tid] + red[64 + tid] + red[128 + tid] + red[192 + tid];
    }
}

__global__ void outmlp_kernel(const float* __restrict__ qstar, const float* __restrict__ W1,
                              const float* __restrict__ b1, const float* __restrict__ W2,
                              const float* __restrict__ b2, float* __restrict__ out) {
    const int g = blockIdx.x, j = threadIdx.x;  // 128 threads
    __shared__ float q[128], red[128];
    q[j] = qstar[g * 128 + j];
    __syncthreads();
    float s = b1[j];
    for (int k = 0; k < 128; ++k) s += q[k] * W1[k * 128 + j];
    s = fmaxf(s, 0.0f);
    red[j] = s * W2[j];
    __syncthreads();
    for (int t = 64; t > 0; t >>= 1) {
        if (j < t) red[j] += red[j + t];
        __syncthreads();
    }
    if (j == 0) out[g] = red[0] + b2[0];
}

// ---------------------------------------------------------------------------
extern "C" void kernel_launch(void* const* d_in, const int* in_sizes, int n_in,
                              void* d_out, int out_size, void* d_ws, size_t ws_size,
                              hipStream_t stream) {
    (void)in_sizes; (void)n_in; (void)out_size; (void)ws_size;
    const float* x      = (const float*)d_in[0];
    const float* eattr  = (const float*)d_in[1];
    const int*   eidx   = (const int*)d_in[2];
    const int*   batch  = (const int*)d_in[3];
    const float* in_W   = (const float*)d_in[4];
    const float* in_b   = (const float*)d_in[5];
    const float* e1_W   = (const float*)d_in[6];
    const float* e1_b   = (const float*)d_in[7];
    const float* e2_W   = (const float*)d_in[8];
    const float* e2_b   = (const float*)d_in[9];
    const float* conv_b = (const float*)d_in[10];
    const float* gWih   = (const float*)d_in[11];
    const float* gWhh   = (const float*)d_in[12];
    const float* gbih   = (const float*)d_in[13];
    const float* gbhh   = (const float*)d_in[14];
    const float* lWih   = (const float*)d_in[15];
    const float* lWhh   = (const float*)d_in[16];
    const float* lbih   = (const float*)d_in[17];
    const float* lbhh   = (const float*)d_in[18];
    const float* o1W    = (const float*)d_in[19];
    const float* o1b    = (const float*)d_in[20];
    const float* o2W    = (const float*)d_in[21];
    const float* o2b    = (const float*)d_in[22];
    float* out = (float*)d_out;
    const int* src = eidx;
    const int* dst = eidx + NE;

    float* ws = (float*)d_ws;
    size_t off = 0;
    auto alloc = [&](size_t n) {
        float* p = ws + off;
        off += (n + 255) & ~(size_t)255;
        return p;
    };
    float* node  = alloc((size_t)NN * 64);      // out == h (shared buffer)
    float* h1    = alloc((size_t)NE * 64);
    float* Bsw   = alloc((size_t)KP_TOT * 4096);
    float* WihT  = alloc(64 * 192);
    float* WhhT  = alloc(64 * 192);
    float* agg   = alloc((size_t)NN * 64);
    float* mbuf  = alloc((size_t)NN * 64);
    float* gi    = alloc((size_t)NN * 192);
    float* gh    = alloc((size_t)NN * 192);
    float* deg   = alloc(NN);
    float* qsc   = alloc((size_t)NG * 128 + NG * 64 + NG * 64);  // qstar|hl|cl
    float* ebuf  = alloc(NN);
    float* qstar = qsc;
    float* hl    = qsc + NG * 128;
    float* cl    = hl + NG * 64;

    // --- prep: swizzled edge-weight matrix, GRU weight transposes, degrees ---
    prep_bsw<<<(KP_TOT * 4096 + 255) / 256, 256, 0, stream>>>(e2_W, e2_b, Bsw);
    transpose_kernel<<<48, 256, 0, stream>>>(gWih, WihT, 192, 64);
    transpose_kernel<<<48, 256, 0, stream>>>(gWhh, WhhT, 192, 64);
    zero_kernel<<<(NN + 255) / 256, 256, 0, stream>>>(deg, NN);
    deg_kernel<<<(NE + 255) / 256, 256, 0, stream>>>(dst, deg);

    // --- input projection & edge MLP layer 1 (WMMA GEMMs) ---
    gemm_wmma<true><<<dim3((NN + 31) / 32, 1), 256, 0, stream>>>(x, in_W, in_b, node, NN, 32, 64);
    gemm_wmma<true><<<dim3((NE + 31) / 32, 1), 256, 0, stream>>>(eattr, e1_W, e1_b, h1, NE, 16, 64);

    // --- NUM_CONV message-passing iterations ---
    for (int t = 0; t < 3; ++t) {
        zero_kernel<<<(NN * 64 + 255) / 256, 256, 0, stream>>>(agg, NN * 64);
        msg_gemm_scatter<<<NE / 64, 256, 0, stream>>>(node, h1, Bsw, src, dst, agg);
        mcalc_kernel<<<(NN * 64 + 255) / 256, 256, 0, stream>>>(agg, deg, conv_b, mbuf);
        gemm_wmma<false><<<dim3((NN + 31) / 32, 3), 256, 0, stream>>>(mbuf, WihT, gbih, gi, NN, 64, 192);
        gemm_wmma<false><<<dim3((NN + 31) / 32, 3), 256, 0, stream>>>(node, WhhT, gbhh, gh, NN, 64, 192);
        gru_gate<<<(NN * 64 + 255) / 256, 256, 0, stream>>>(gi, gh, node);
    }

    // --- Set2Set pooling ---
    zero_kernel<<<(NG * 256 + 255) / 256, 256, 0, stream>>>(qsc, NG * 256);
    for (int t = 0; t < 3; ++t) {
        lstm_kernel<<<NG, 256, 0, stream>>>(qstar, hl, cl, lWih, lWhh, lbih, lbhh);
        attn_kernel<<<NG, 256, 0, stream>>>(node, batch, hl, qstar, ebuf);
    }

    // --- output MLP ---
    outmlp_kernel<<<NG, 128, 0, stream>>>(qstar, o1W, o1b, o2W, o2b, out);
}